// SelMul_functional_81647328297698
// MI455X (gfx1250) — compile-verified
//
#include <hip/hip_runtime.h>

typedef __attribute__((ext_vector_type(2))) float v2f;
typedef __attribute__((ext_vector_type(8))) float v8f;

#define N_COLS 1024
#define P_OUT  524800          // 1024*1025/2 outputs per batch row

// One wave computes one 16x16 output tile (i-tile ti, j-tile tj, ti<=tj) of the
// per-row outer product via a rank-1 V_WMMA_F32_16X16X4_F32, then writes the
// upper-triangular elements with coalesced non-temporal dword stores.
//
// Grid: x = tj-group (8 waves/block cover tj = x*8+wave), y = ti (64), z = b.
// Waves with tj < ti exit immediately (wave-uniform branch, ~8 instrs).
__global__ __launch_bounds__(256)
void selmul_wmma_kernel(const float* __restrict__ x, float* __restrict__ out) {
    const int lane = threadIdx.x & 31;
    const int wave = threadIdx.x >> 5;
    const int ti   = blockIdx.y;
    const int tj   = blockIdx.x * 8 + wave;
    const int b    = blockIdx.z;

    if (tj < ti) return;   // uniform per wave: survivors keep EXEC all-ones

    // Load the two 16-element row segments (input is L2-resident).
    const int  l16 = lane & 15;
    const bool lo  = lane < 16;
    const float av = x[b * N_COLS + ti * 16 + l16];
    const float bv = x[b * N_COLS + tj * 16 + l16];

    // A (16x4 f32): K=0 lives in VGPR0 of lanes 0-15; K=1..3 zeroed.
    // B (4x16 f32): row K=0 lives in lanes 0-15 of VGPR0; K=1..3 zeroed.
    v2f A; A.x = lo ? av : 0.0f; A.y = 0.0f;
    v2f B; B.x = lo ? bv : 0.0f; B.y = 0.0f;
    v8f C = {};

    // D = A x B + 0  ->  16x16 outer product a * b^T in one matrix op.
    C = __builtin_amdgcn_wmma_f32_16x16x4_f32(
            /*neg_a=*/false, A, /*neg_b=*/false, B,
            /*c_mod=*/(short)0, C, /*reuse_a=*/false, /*reuse_b=*/false);

    // D layout: VGPR r -> M = r + 8*(lane/16), N = lane%16.
    const int i0 = ti * 16 + ((lane >> 4) << 3);     // first i this lane stores
    const int j  = tj * 16 + l16;

    // off(i,j) = b*P + i*(2047-i)/2 + j ; row-to-row delta is (1023 - i).
    unsigned off = (unsigned)b * (unsigned)P_OUT
                 + (unsigned)(i0 * (2 * N_COLS - 1 - i0)) / 2u
                 + (unsigned)j;

    if (ti != tj) {
        // Fully valid tile: unpredicated streaming stores (store + add only).
        #pragma unroll
        for (int r = 0; r < 8; ++r) {
            __builtin_nontemporal_store(C[r], out + off);
            off += (unsigned)(N_COLS - 1 - (i0 + r));
        }
    } else {
        // Diagonal tile: keep only j >= i elements.
        #pragma unroll
        for (int r = 0; r < 8; ++r) {
            if (j >= i0 + r) __builtin_nontemporal_store(C[r], out + off);
            off += (unsigned)(N_COLS - 1 - (i0 + r));
        }
    }
}

extern "C" void kernel_launch(void* const* d_in, const int* in_sizes, int n_in,
                              void* d_out, int out_size, void* d_ws, size_t ws_size,
                              hipStream_t stream) {
    const float* x = (const float*)d_in[0];
    float* out     = (float*)d_out;
    const int B    = in_sizes[0] / N_COLS;     // 256 for the reference shapes

    dim3 block(256, 1, 1);                     // 8 waves -> 8 tj tiles per block
    dim3 grid(8, 64, B);                       // (tj/8, ti, batch)
    selmul_wmma_kernel<<<grid, block, 0, stream>>>(x, out);
}